// GbROptim_75763223101516
// MI455X (gfx1250) — compile-verified
//
#include <hip/hip_runtime.h>
#include <hip/hip_bf16.h>
#include <math.h>

typedef __attribute__((ext_vector_type(2))) float v2f;
typedef __attribute__((ext_vector_type(8))) float v8f;

#define EMA_   0.999f
#define LOWER_ (1.0f/3.0f)
#define UPPER_ 5.0f

// ---------------------------------------------------------------- zero scratch
__global__ void zero_kernel(float* p, int n) {
    int i = blockIdx.x * blockDim.x + threadIdx.x;
    if (i < n) p[i] = 0.0f;
}

// ------------------------------------------------- cw2 = softmax(logits)*C etc.
__global__ void prep_kernel(const float* __restrict__ logits,
                            const float* __restrict__ cwbuf,
                            float* __restrict__ cw2, float* __restrict__ cwclip,
                            float* __restrict__ out_cw2, int C) {
    __shared__ float red[256];
    int tid = threadIdx.x;
    float m = -3.402823466e38f;
    for (int j = tid; j < C; j += 256) m = fmaxf(m, logits[j]);
    red[tid] = m; __syncthreads();
    for (int s = 128; s > 0; s >>= 1) { if (tid < s) red[tid] = fmaxf(red[tid], red[tid+s]); __syncthreads(); }
    m = red[0]; __syncthreads();
    float sum = 0.0f;
    for (int j = tid; j < C; j += 256) sum += __expf(logits[j] - m);
    red[tid] = sum; __syncthreads();
    for (int s = 128; s > 0; s >>= 1) { if (tid < s) red[tid] += red[tid+s]; __syncthreads(); }
    sum = red[0];
    float inv = (float)C / sum;
    for (int j = tid; j < C; j += 256) {
        float v = __expf(logits[j] - m) * inv;
        cw2[j] = v;
        out_cw2[j] = v;
        cwclip[j] = fminf(fmaxf(cwbuf[j], LOWER_), UPPER_);
    }
}

// ------------------- per-row softmax stats + per-row scalars (1 wave per row)
__global__ void row_kernel(const float* __restrict__ cs, const int* __restrict__ label,
                           const float* __restrict__ weight,
                           const float* __restrict__ cw2, const float* __restrict__ cwclip,
                           float* __restrict__ logZ_out, float* __restrict__ u_out,
                           float* __restrict__ posacc, float* __restrict__ diagacc,
                           float* __restrict__ scal, int N, int C) {
    int lane = threadIdx.x & 31;
    int wid  = threadIdx.x >> 5;
    int row  = blockIdx.x * 8 + wid;
    __shared__ float sA[8], sB[8], sU[8];
    float a_c = 0.0f, b_c = 0.0f, u_c = 0.0f;
    if (row < N) {
        const float* rp = cs + (size_t)row * C;
        float m = -3.402823466e38f;
        for (int j = lane; j < C; j += 32) m = fmaxf(m, rp[j]);
        for (int o = 16; o >= 1; o >>= 1) m = fmaxf(m, __shfl_xor(m, o, 32));
        float s = 0.0f;
        for (int j = lane; j < C; j += 32) s += __expf(rp[j] - m);
        for (int o = 16; o >= 1; o >>= 1) s += __shfl_xor(s, o, 32);
        float lZ = m + __logf(s);
        if (lane == 0) {
            int   l   = label[row];
            float csl = rp[l];
            float CE  = lZ - csl;          // -log_softmax at label
            float p   = __expf(-CE);
            float w   = weight[row];
            float pl  = __expf(csl - lZ) + 1e-5f;            // prob (+eps) at label
            float fg  = w * ((1.0f - pl) / pl - __logf(pl)); // focal_grad at label
            float s_n = fg * pl;                             // a[n,label]
            float u_n = cw2[l] * s_n;
            logZ_out[row] = lZ;
            u_out[row]    = u_n;
            atomicAdd(&posacc[l],  (1.0f - pl) * pl * fg);
            atomicAdd(&diagacc[l], s_n * pl);
            float cwl = cwclip[l];
            a_c = cwl;                           // -> Σ cw[label]  (for wm)
            b_c = (1.0f - p) * CE * cwl * w;     // -> loss numerator
            u_c = u_n;                           // -> Σ u  (epsilon correction)
        }
    }
    if (lane == 0) { sA[wid] = a_c; sB[wid] = b_c; sU[wid] = u_c; }
    __syncthreads();
    if (threadIdx.x == 0) {
        float ta = 0, tb = 0, tu = 0;
        for (int i = 0; i < 8; i++) { ta += sA[i]; tb += sB[i]; tu += sU[i]; }
        atomicAdd(&scal[0], ta); atomicAdd(&scal[1], tb); atomicAdd(&scal[2], tu);
    }
}

// ------- vec1[j] = Σ_n u_n * exp(cs[n,j]-logZ[n])  via fp32 WMMA 16x16x4
// A tile 16x4: row 0 = u[n0..n0+3] (lanes 0/16 carry K pairs per ISA layout),
// B tile 4x16: exp values. Branchless masking (EXEC stays all-ones at WMMA),
// 2x unrolled with independent accumulators to break the WMMA RAW chain.
__global__ void vec1_kernel(const float* __restrict__ cs, const float* __restrict__ logZ,
                            const float* __restrict__ uu, float* __restrict__ vec1,
                            int N, int C, int slabRows) {
    int lane  = threadIdx.x;            // blockDim.x == 32, one wave
    int j0    = blockIdx.x * 16;
    int col   = j0 + (lane & 15);
    bool colOK = (col < C);
    int   colC  = colOK ? col : 0;      // safe column index
    float bmask = colOK ? 1.0f : 0.0f;  // kill OOB columns arithmetically
    float amask = ((lane & 15) == 0) ? 1.0f : 0.0f;  // A row 0 only
    int nStart = blockIdx.y * slabRows;
    int nEnd   = nStart + slabRows; if (nEnd > N) nEnd = N;
    int kb = (lane >> 4) * 2;           // K pair for this half-wave: {0,1} or {2,3}

    v8f acc0 = {};
    v8f acc1 = {};
    int n0 = nStart;
    for (; n0 + 7 < nEnd; n0 += 8) {
        // ---- k-step A: rows n0..n0+3
        {
            int na = n0 + kb;
            v2f a; a[0] = uu[na] * amask; a[1] = uu[na + 1] * amask;
            float lz0 = logZ[na], lz1 = logZ[na + 1];
            const float* r0 = cs + (size_t)na * C;
            v2f b;
            b[0] = __expf(r0[colC]     - lz0) * bmask;
            b[1] = __expf(r0[C + colC] - lz1) * bmask;
            acc0 = __builtin_amdgcn_wmma_f32_16x16x4_f32(
                       false, a, false, b, (short)0, acc0, false, false);
        }
        // ---- k-step B: rows n0+4..n0+7 (independent accumulator)
        {
            int na = n0 + 4 + kb;
            v2f a; a[0] = uu[na] * amask; a[1] = uu[na + 1] * amask;
            float lz0 = logZ[na], lz1 = logZ[na + 1];
            const float* r0 = cs + (size_t)na * C;
            v2f b;
            b[0] = __expf(r0[colC]     - lz0) * bmask;
            b[1] = __expf(r0[C + colC] - lz1) * bmask;
            acc1 = __builtin_amdgcn_wmma_f32_16x16x4_f32(
                       false, a, false, b, (short)0, acc1, false, false);
        }
    }
    for (; n0 + 3 < nEnd; n0 += 4) {    // tail (slab not multiple of 8)
        int na = n0 + kb;
        v2f a; a[0] = uu[na] * amask; a[1] = uu[na + 1] * amask;
        float lz0 = logZ[na], lz1 = logZ[na + 1];
        const float* r0 = cs + (size_t)na * C;
        v2f b;
        b[0] = __expf(r0[colC]     - lz0) * bmask;
        b[1] = __expf(r0[C + colC] - lz1) * bmask;
        acc0 = __builtin_amdgcn_wmma_f32_16x16x4_f32(
                   false, a, false, b, (short)0, acc0, false, false);
    }
    float r = acc0[0] + acc1[0];        // C[0, lane] partial
    if (lane < 16 && colOK) atomicAdd(&vec1[col], r);
}

// --------------- colsum[j] = Σ_i cw2[i] * cga[i,j]  (split-K over blockIdx.y)
__global__ void cga_kernel(const float* __restrict__ cga, const float* __restrict__ cw2,
                           float* __restrict__ colsum, int C) {
    int j = blockIdx.x * blockDim.x + threadIdx.x;
    if (j >= C) return;
    int chunk = (C + gridDim.y - 1) / gridDim.y;
    int i0 = blockIdx.y * chunk;
    int i1 = i0 + chunk; if (i1 > C) i1 = C;
    float s = 0.0f;
    for (int i = i0; i < i1; i++) s += cw2[i] * cga[(size_t)i * C + j];
    atomicAdd(&colsum[j], s);
}

// ---------------------------------------------------------- final combination
__global__ void final_kernel(const float* __restrict__ cga, const float* __restrict__ logits,
                             const float* __restrict__ wmean, const float* __restrict__ cw2,
                             const float* __restrict__ posacc, const float* __restrict__ diagacc,
                             const float* __restrict__ vec1, const float* __restrict__ colsum,
                             const float* __restrict__ scal, float* __restrict__ tcw,
                             float* __restrict__ out, int N, int C) {
    __shared__ float red[256];
    int tid = threadIdx.x;
    float sumA = scal[0], sumB = scal[1], sumU = scal[2];
    float invN = 1.0f / (float)N;
    for (int j = tid; j < C; j += 256) {
        float pos   = posacc[j]  * invN;                        // pos_grad[j]
        float cdia  = diagacc[j] * invN;                        // corr[j,j] (pre-replace)
        float raw   = 100.0f * (vec1[j] + 1e-5f * sumU) * invN; // Σ_i cw2_i*100*corr[i,j]
        float c2    = cw2[j];
        float corrsum = raw - c2 * (100.0f * cdia) + c2 * (-100.0f * pos); // diag replaced
        float acc_col = EMA_ * colsum[j] + (1.0f - EMA_) * corrsum;  // Σ_i cw2_i*accu[i,j]
        float acc_dia = EMA_ * cga[(size_t)j * C + j] + (1.0f - EMA_) * (-100.0f * pos);
        tcw[j] = acc_col / (-acc_dia + 1e-6f) + c2;
    }
    __syncthreads();
    float local = 0.0f;
    for (int j = tid; j < C - 1; j += 256) local += tcw[j];
    red[tid] = local; __syncthreads();
    for (int s = 128; s > 0; s >>= 1) { if (tid < s) red[tid] += red[tid+s]; __syncthreads(); }
    float S = red[0]; __syncthreads();
    local = 0.0f;
    for (int j = tid; j < C; j += 256) {
        float pre = (j == C - 1) ? ((float)C - S) : tcw[j];
        float hi  = (j == C - 1) ? 1.0f : UPPER_;   // clip(L,U) then clip(L,1) == clip(L,1)
        float val = fminf(fmaxf(pre, LOWER_), hi);
        float d   = logf(val) - logits[j];
        local += d * d;
    }
    red[tid] = local; __syncthreads();
    for (int s = 128; s > 0; s >>= 1) { if (tid < s) red[tid] += red[tid+s]; __syncthreads(); }
    if (tid == 0) {
        float wm = EMA_ * wmean[0] + (1.0f - EMA_) * (sumA * invN);
        out[0] = (sumB * invN) / wm;   // loss (SELF_LOSS_WEIGHT = 1)
        out[1] = red[0] / (float)C;    // loss_grad
    }
}

extern "C" void kernel_launch(void* const* d_in, const int* in_sizes, int n_in,
                              void* d_out, int out_size, void* d_ws, size_t ws_size,
                              hipStream_t stream) {
    const float* cs     = (const float*)d_in[0];
    const int*   label  = (const int*)  d_in[1];
    const float* weight = (const float*)d_in[2];
    const float* logits = (const float*)d_in[3];
    const float* cwbuf  = (const float*)d_in[4];
    const float* cga    = (const float*)d_in[5];
    const float* wmean  = (const float*)d_in[6];
    int N  = in_sizes[1];
    int C  = in_sizes[3];
    int Cp = ((C + 15) / 16) * 16;

    float* w       = (float*)d_ws;
    float* cw2     = w;
    float* cwclip  = w + 1 * Cp;
    float* posacc  = w + 2 * Cp;
    float* diagacc = w + 3 * Cp;
    float* vec1    = w + 4 * Cp;
    float* colsum  = w + 5 * Cp;
    float* scal    = w + 6 * Cp;        // [0]=sumA [1]=sumB [2]=sumU
    float* tcw     = w + 6 * Cp + 16;
    float* logZ    = w + 7 * Cp + 16;
    float* uu      = logZ + N;
    float* out     = (float*)d_out;

    int zn = 4 * Cp + 16;               // posacc..scal inclusive
    zero_kernel<<<(zn + 255) / 256, 256, 0, stream>>>(posacc, zn);
    prep_kernel<<<1, 256, 0, stream>>>(logits, cwbuf, cw2, cwclip, out + 2, C);
    row_kernel<<<(N + 7) / 8, 256, 0, stream>>>(cs, label, weight, cw2, cwclip,
                                                logZ, uu, posacc, diagacc, scal, N, C);
    int slabs = 32;
    int slabRows = (N + slabs - 1) / slabs;   // N multiple of 256 -> 512, multiple of 8
    dim3 g1((C + 15) / 16, slabs);
    vec1_kernel<<<g1, 32, 0, stream>>>(cs, logZ, uu, vec1, N, C, slabRows);
    dim3 g2((C + 255) / 256, 8);
    cga_kernel<<<g2, 256, 0, stream>>>(cga, cw2, colsum, C);
    final_kernel<<<1, 256, 0, stream>>>(cga, logits, wmean, cw2, posacc, diagacc,
                                        vec1, colsum, scal, tcw, out, N, C);
}